// CTCLayer_44521630991170
// MI455X (gfx1250) — compile-verified
//
#include <hip/hip_runtime.h>
#include <hip/hip_bf16.h>

#define CTC_EPS 1e-7f
#define CTC_NEG -1e30f

// logaddexp identical in spirit to jnp.logaddexp; safe for -1e30 sentinels.
__device__ __forceinline__ float lae(float a, float b) {
    float mx = fmaxf(a, b);
    float mn = fminf(a, b);
    return mx + log1pf(__expf(mn - mx));
}

// One async DMA: 32 lanes x 16B = one 512B probability row into LDS.
// Non-temporal: y_pred is streamed exactly once (131 MB), no reuse in L2.
__device__ __forceinline__ void async_row(unsigned lds_addr, unsigned byte_off,
                                          const float* sbase) {
    asm volatile("global_load_async_to_lds_b128 %0, %1, %2 th:TH_LOAD_NT"
                 :
                 : "v"(lds_addr), "v"(byte_off), "s"(sbase)
                 : "memory");
}

// Wait until the oldest outstanding row (the one we need now) has landed.
// `rem` = number of rows currently outstanding (completions are in-order).
__device__ __forceinline__ void wait_oldest(int rem) {
    if (rem >= 3)      asm volatile("s_wait_asynccnt 2" ::: "memory");
    else if (rem == 2) asm volatile("s_wait_asynccnt 1" ::: "memory");
    else               asm volatile("s_wait_asynccnt 0" ::: "memory");
}

__global__ __launch_bounds__(512)
void ctc_batch_cost_kernel(const float* __restrict__ y_pred,   // [B,T,C]
                           const int*   __restrict__ labels,   // [B,L]
                           const int*   __restrict__ feat_lens,// [B]
                           const int*   __restrict__ label_lens,//[B]
                           float*       __restrict__ out,      // [B]
                           int T, int L) {
    constexpr int C = 128;
    constexpr unsigned ROWB = C * 4u;    // 512 bytes per probability row
    const int b     = blockIdx.x;
    const int tid   = threadIdx.x;
    const int S     = 2 * L + 1;         // 401
    const int blank = C - 1;

    __shared__ float abuf[2][514];       // alpha double buffer, +2 NEG padding
    __shared__ float prob[4][C];         // 4-slot ring of async-DMA'd rows

    // CTCLayer subsampling: twice ceil-div by 2
    int fl = feat_lens[b];
    fl = (fl + 1) >> 1;
    fl = (fl + 1) >> 1;
    if (fl > T) fl = T;
    if (fl < 1) fl = 1;

    const float* base = y_pred + (size_t)b * (size_t)T * (size_t)C;

    // per-state extended-label class and skip_ok flag
    int  cls  = blank;
    bool skip = false;
    if (tid < S && (tid & 1)) {
        cls = labels[(size_t)b * L + ((tid - 1) >> 1)];
        if (tid >= 3) {
            int prev = labels[(size_t)b * L + ((tid - 3) >> 1)];
            skip = (cls != prev);   // ext[s] != blank guaranteed (labels < C-1)
        }
    }

    const unsigned lane = tid & 31u;
    // low 32 bits of the flat pointer == wave-relative LDS byte address
    const unsigned lds_prob = (unsigned)(unsigned long long)(&prob[0][0]);

    // ---- prologue: prefetch rows 0..2 (depth-3 pipeline, 4-slot ring) ----
    if (tid < 32) {
        for (int r = 0; r < 3; ++r) {
            if (r < fl) {
                async_row(lds_prob + (unsigned)(r & 3) * ROWB + lane * 16u,
                          (unsigned)r * ROWB + lane * 16u, base);
            }
        }
    }
    // NEG padding for alpha[-1], alpha[-2] in both buffers (never overwritten)
    if (tid < 2) { abuf[0][tid] = CTC_NEG; abuf[1][tid] = CTC_NEG; }

    if (tid < 32) wait_oldest(fl < 3 ? fl : 3);   // row 0 landed
    __syncthreads();

    // t = 0 init: alpha0[s] = emit0[s] for s in {0,1}, NEG otherwise.
    // Input to step t=1 lives in abuf[1].
    if (tid < S) {
        float a = CTC_NEG;
        if (tid < 2) a = __logf(prob[0][cls] + CTC_EPS);
        abuf[1][tid + 2] = a;
    }
    // keep pipeline full: issue row 3 (slot 3, untouched so far)
    if (tid < 32 && 3 < fl) {
        async_row(lds_prob + 3u * ROWB + lane * 16u,
                  3u * ROWB + lane * 16u, base);
    }

    for (int t = 1; t < fl; ++t) {
        // outstanding rows at this point: {t .. min(t+2, fl-1)}
        if (tid < 32) {
            int rem = fl - t;           // >= 1
            wait_oldest(rem);           // row t landed in slot t&3
        }
        __syncthreads();                // + prev step's alpha visible
        const int p = t & 1;
        // prefetch row t+3 into slot (t+3)&3 == (t-1)&3, finished last step
        if (tid < 32 && (t + 3) < fl) {
            unsigned slot = (unsigned)((t + 3) & 3);
            async_row(lds_prob + slot * ROWB + lane * 16u,
                      (unsigned)(t + 3) * ROWB + lane * 16u, base);
        }
        if (tid < S) {
            float e  = __logf(prob[t & 3][cls] + CTC_EPS);
            float a0 = abuf[p][tid + 2];
            float a1 = abuf[p][tid + 1];
            float a2 = abuf[p][tid];
            float m  = lae(a0, a1);
            if (skip) m = lae(m, a2);
            abuf[p ^ 1][tid + 2] = m + e;
        }
    }
    __syncthreads();

    if (tid == 0) {
        int ll = label_lens[b];
        int i1 = 2 * ll - 1;
        int i2 = 2 * ll;
        const float* af = abuf[fl & 1];       // alpha after step fl-1
        out[b] = -lae(af[i1 + 2], af[i2 + 2]);
    }
}

extern "C" void kernel_launch(void* const* d_in, const int* in_sizes, int n_in,
                              void* d_out, int out_size, void* d_ws, size_t ws_size,
                              hipStream_t stream) {
    const float* y_pred     = (const float*)d_in[0];
    const int*   labels     = (const int*)d_in[1];
    const int*   feat_lens  = (const int*)d_in[2];
    const int*   label_lens = (const int*)d_in[3];
    float*       out        = (float*)d_out;

    const int B = in_sizes[2];                  // 128
    const int L = in_sizes[1] / B;              // 200
    const int C = 128;
    const int T = in_sizes[0] / (B * C);        // 2000

    int S = 2 * L + 1;                          // 401
    int block = ((S + 31) / 32) * 32;           // 416 = 13 wave32s
    if (block < 64)  block = 64;
    if (block > 512) block = 512;

    ctc_batch_cost_kernel<<<B, block, 0, stream>>>(y_pred, labels, feat_lens,
                                                   label_lens, out, T, L);
}